// MultiDilatelocalAttention_32418413150926
// MI455X (gfx1250) — compile-verified
//
#include <hip/hip_runtime.h>

typedef __attribute__((ext_vector_type(2))) float v2f;
typedef __attribute__((ext_vector_type(8))) float v8f;

#define DIMC 192
#define HW   16384
#define HD   24
#define NB   4

// ---------------------------------------------------------------------------
// QKV GEMM: qkv[b, o, p] = sum_c Wqkv[o, c] * x[b, c, p]
//   M = 576 (36 tiles), K = 192 (48 steps of 4), N = 16384 (256 tiles of 64).
//   Each wave computes a 16x64 tile: 4 accumulators share one A fetch per
//   K-step -> 4 v_wmma_f32_16x16x4_f32 per 9 loads.
// A layout (16x4 f32): lanes 0-15 hold K={0,1}, lanes 16-31 hold K={2,3}.
// B layout (4x16 f32): VGPR0/1 = rows K(+0)/K(+1), lane halves K / K+2.
// C/D layout: VGPR r -> row m0+r (lanes 0-15) / m0+8+r (lanes 16-31).
// ---------------------------------------------------------------------------
__global__ __launch_bounds__(256) void qkv_gemm_kernel(
    const float* __restrict__ Wqkv, const float* __restrict__ x,
    float* __restrict__ qkv) {
  const int lane = threadIdx.x & 31;
  const int wid  = (blockIdx.x << 3) + (threadIdx.x >> 5);
  const int per_batch = 256 * 36;                  // n-tiles * m-tiles
  const int b   = wid / per_batch;
  const int rem = wid - b * per_batch;
  const int nt  = rem / 36;                        // 36 consecutive waves share nt; a block may span two
  const int mt  = rem - nt * 36;
  const int m0 = mt << 4, p0 = nt << 6;
  const int lh = lane & 15;
  const int kh = (lane >> 4) << 1;                 // 0 (lanes 0-15) or 2 (16-31)

  const float* wrow = Wqkv + (size_t)(m0 + lh) * DIMC + kh;
  const float* xcol = x + ((size_t)b * DIMC + kh) * HW + p0 + lh;
  __builtin_prefetch(wrow, 0, 3);                  // global_prefetch_b8 (weights -> L2/L0)

  v8f acc0 = {}, acc1 = {}, acc2 = {}, acc3 = {};
#pragma unroll 4
  for (int k = 0; k < DIMC; k += 4) {
    v2f a = *(const v2f*)(wrow + k);               // b64: W[m, k+kh], W[m, k+kh+1]
    const float* c0 = xcol + (size_t)k * HW;
    const float* c1 = xcol + (size_t)(k + 1) * HW;
    v2f b0, b1, b2, b3;
    b0.x = c0[0];  b0.y = c1[0];
    b1.x = c0[16]; b1.y = c1[16];
    b2.x = c0[32]; b2.y = c1[32];
    b3.x = c0[48]; b3.y = c1[48];
    acc0 = __builtin_amdgcn_wmma_f32_16x16x4_f32(false, a, false, b0, (short)0, acc0, false, false);
    acc1 = __builtin_amdgcn_wmma_f32_16x16x4_f32(false, a, false, b1, (short)0, acc1, false, false);
    acc2 = __builtin_amdgcn_wmma_f32_16x16x4_f32(false, a, false, b2, (short)0, acc2, false, false);
    acc3 = __builtin_amdgcn_wmma_f32_16x16x4_f32(false, a, false, b3, (short)0, acc3, false, false);
  }

  float* out = qkv + (size_t)b * 576 * HW + p0 + lh;
  const int mbase = m0 + ((lane >> 4) << 3);
#pragma unroll
  for (int r = 0; r < 8; ++r) {
    float* row = out + (size_t)(mbase + r) * HW;
    row[0]  = acc0[r];
    row[16] = acc1[r];
    row[32] = acc2[r];
    row[48] = acc3[r];
  }
}

// ---------------------------------------------------------------------------
// Dilated local attention. qkv is (B, 576, 16384) with o = ((s*2+i)*4+j)*24+t
// (s=q/k/v, i=dilation idx, j=head-in-group, t=channel). Output y is the
// reference's concat(axis=0) flat layout: row = (i*16 + b*4 + j)*24 + t,
// i.e. y is a flat (768, 16384) tensor later reinterpreted as (B, 192, HW).
// Softmax runs over all 9 taps; zero-padded taps contribute logit 0 (exact
// match to unfold's zero padding of k), padded v contributes nothing.
// ---------------------------------------------------------------------------
__global__ __launch_bounds__(256) void attn_kernel(
    const float* __restrict__ qkv, float* __restrict__ y) {
  const int p = blockIdx.x * 256 + threadIdx.x;    // pixel 0..16383
  const int r = blockIdx.y;                        // 0..31
  const int i = r >> 4, b = (r >> 2) & 3, j = r & 3;
  const int dil = i ? 3 : 2;                       // DILS = (2, 3)
  const int h = p >> 7, w = p & 127;
  const float scale = 0.2041241452319315f;         // 24^-0.5

  const float* Bq = qkv + (size_t)b * 576 * HW;
  const int oq = ((0 * 2 + i) * 4 + j) * HD;
  const int ok = ((1 * 2 + i) * 4 + j) * HD;
  const int ov = ((2 * 2 + i) * 4 + j) * HD;

  float q[HD];
#pragma unroll
  for (int t = 0; t < HD; ++t) q[t] = Bq[(size_t)(oq + t) * HW + p];

  float lg[9];
  int   pw[9];
#pragma unroll
  for (int kk = 0; kk < 9; ++kk) {
    const int hh = h + (kk / 3 - 1) * dil;
    const int ww = w + (kk % 3 - 1) * dil;
    const bool in = (hh >= 0) && (hh < 128) && (ww >= 0) && (ww < 128);
    const int pp = (hh << 7) + ww;
    pw[kk] = in ? pp : -1;
    float acc = 0.f;
    if (in) {
#pragma unroll
      for (int t = 0; t < HD; ++t)
        acc = fmaf(q[t], Bq[(size_t)(ok + t) * HW + pp], acc);
    }
    lg[kk] = acc * scale;
  }

  float mx = lg[0];
#pragma unroll
  for (int kk = 1; kk < 9; ++kk) mx = fmaxf(mx, lg[kk]);
  float s = 0.f;
#pragma unroll
  for (int kk = 0; kk < 9; ++kk) { lg[kk] = __expf(lg[kk] - mx); s += lg[kk]; }
  const float inv = 1.f / s;

  float o[HD];
#pragma unroll
  for (int t = 0; t < HD; ++t) o[t] = 0.f;
#pragma unroll
  for (int kk = 0; kk < 9; ++kk) {
    if (pw[kk] >= 0) {
      const float a = lg[kk];
      const int pp = pw[kk];
#pragma unroll
      for (int t = 0; t < HD; ++t)
        o[t] = fmaf(a, Bq[(size_t)(ov + t) * HW + pp], o[t]);
    }
  }

  float* Y = y + (size_t)r * HD * HW + p;
#pragma unroll
  for (int t = 0; t < HD; ++t) Y[(size_t)t * HW] = o[t] * inv;
}

// ---------------------------------------------------------------------------
// Proj GEMM + bias: out[b, o, p] = sum_c Wproj[o, c] * y[b*192 + c, p] + bproj[o]
//   M = 192 (12 tiles), K = 192, N = 16384 (256 tiles of 64).
// ---------------------------------------------------------------------------
__global__ __launch_bounds__(256) void proj_gemm_kernel(
    const float* __restrict__ Wproj, const float* __restrict__ bproj,
    const float* __restrict__ y, float* __restrict__ out) {
  const int lane = threadIdx.x & 31;
  const int wid  = (blockIdx.x << 3) + (threadIdx.x >> 5);
  const int per_batch = 256 * 12;
  const int b   = wid / per_batch;
  const int rem = wid - b * per_batch;
  const int nt  = rem / 12;
  const int mt  = rem - nt * 12;
  const int m0 = mt << 4, p0 = nt << 6;
  const int lh = lane & 15;
  const int kh = (lane >> 4) << 1;

  const float* wrow = Wproj + (size_t)(m0 + lh) * DIMC + kh;
  const float* ycol = y + ((size_t)b * DIMC + kh) * HW + p0 + lh;
  __builtin_prefetch(wrow, 0, 3);

  v8f acc0 = {}, acc1 = {}, acc2 = {}, acc3 = {};
#pragma unroll 4
  for (int k = 0; k < DIMC; k += 4) {
    v2f a = *(const v2f*)(wrow + k);
    const float* c0 = ycol + (size_t)k * HW;
    const float* c1 = ycol + (size_t)(k + 1) * HW;
    v2f b0, b1, b2, b3;
    b0.x = c0[0];  b0.y = c1[0];
    b1.x = c0[16]; b1.y = c1[16];
    b2.x = c0[32]; b2.y = c1[32];
    b3.x = c0[48]; b3.y = c1[48];
    acc0 = __builtin_amdgcn_wmma_f32_16x16x4_f32(false, a, false, b0, (short)0, acc0, false, false);
    acc1 = __builtin_amdgcn_wmma_f32_16x16x4_f32(false, a, false, b1, (short)0, acc1, false, false);
    acc2 = __builtin_amdgcn_wmma_f32_16x16x4_f32(false, a, false, b2, (short)0, acc2, false, false);
    acc3 = __builtin_amdgcn_wmma_f32_16x16x4_f32(false, a, false, b3, (short)0, acc3, false, false);
  }

  float* O = out + (size_t)b * DIMC * HW + p0 + lh;
  const int mbase = m0 + ((lane >> 4) << 3);
#pragma unroll
  for (int r = 0; r < 8; ++r) {
    const float bias = bproj[mbase + r];
    float* row = O + (size_t)(mbase + r) * HW;
    row[0]  = acc0[r] + bias;
    row[16] = acc1[r] + bias;
    row[32] = acc2[r] + bias;
    row[48] = acc3[r] + bias;
  }
}

// ---------------------------------------------------------------------------
extern "C" void kernel_launch(void* const* d_in, const int* in_sizes, int n_in,
                              void* d_out, int out_size, void* d_ws, size_t ws_size,
                              hipStream_t stream) {
  (void)in_sizes; (void)n_in; (void)out_size; (void)ws_size;
  const float* x     = (const float*)d_in[0];
  const float* Wqkv  = (const float*)d_in[1];
  const float* Wproj = (const float*)d_in[2];
  const float* bproj = (const float*)d_in[3];
  float* out = (float*)d_out;

  float* qkv = (float*)d_ws;                       // (4, 576, 16384) f32 ~151 MB
  float* y   = qkv + (size_t)NB * 576 * HW;        // (768, 16384)    f32  ~50 MB

  // QKV GEMM: 4 * 36 * 256 wave-tiles (16x64 each), 8 waves per block
  qkv_gemm_kernel<<<(NB * 36 * 256) / 8, 256, 0, stream>>>(Wqkv, x, qkv);

  // Attention: one thread per (pixel, head-instance)
  attn_kernel<<<dim3(HW / 256, 32), 256, 0, stream>>>(qkv, y);

  // Proj GEMM + bias: 4 * 12 * 256 wave-tiles
  proj_gemm_kernel<<<(NB * 12 * 256) / 8, 256, 0, stream>>>(Wproj, bproj, y, out);
}